// GCNEncoder_65300682769036
// MI455X (gfx1250) — compile-verified
//
#include <hip/hip_runtime.h>
#include <hip/hip_bf16.h>
#include <math.h>

// ---------------- problem constants (match reference) ----------------
#define N_NODES   50000
#define N_EDGES   800000
#define C_DIM     128
#define EDGE_DIM  8
#define N_LAYERS  3
#define M_TILES   (N_NODES / 16)   // 3125, exact

typedef __attribute__((ext_vector_type(2))) float v2f;
typedef __attribute__((ext_vector_type(8))) float v8f;

// ---------------------------------------------------------------------
// Kernel 1: per-edge scalar weight  w[e] = softplus(edge_attr[e,:] . ew + eb)
// ---------------------------------------------------------------------
__global__ void __launch_bounds__(256)
edge_weight_kernel(const float* __restrict__ edge_attr,
                   const float* __restrict__ ew,   // [8]
                   const float* __restrict__ eb,   // [1]
                   float* __restrict__ w, int nE)
{
    int e = blockIdx.x * blockDim.x + threadIdx.x;
    if (e >= nE) return;
    const float4* ea = (const float4*)(edge_attr + (size_t)e * EDGE_DIM);
    float4 a0 = ea[0];
    float4 a1 = ea[1];
    float z = a0.x * ew[0] + a0.y * ew[1] + a0.z * ew[2] + a0.w * ew[3]
            + a1.x * ew[4] + a1.y * ew[5] + a1.z * ew[6] + a1.w * ew[7]
            + eb[0];
    // numerically stable softplus
    float sp;
    if (z > 20.0f)        sp = z;
    else if (z < -20.0f)  sp = expf(z);
    else                  sp = log1pf(expf(z));
    w[e] = sp;
}

// ---------------------------------------------------------------------
// Kernel 2: zero the aggregation buffer (float4 stores)
// ---------------------------------------------------------------------
__global__ void __launch_bounds__(256)
zero_kernel(float4* __restrict__ p, int n4)
{
    int i = blockIdx.x * blockDim.x + threadIdx.x;
    if (i < n4) p[i] = make_float4(0.f, 0.f, 0.f, 0.f);
}

// ---------------------------------------------------------------------
// Kernel 3: wave-per-edge message + scatter-add.
// Each 32-lane wave handles one edge: lane L covers channels [4L, 4L+4).
// Row gathers are fully coalesced 512B reads; x/agg live in the 192MB L2.
// ---------------------------------------------------------------------
__global__ void __launch_bounds__(256)
scatter_kernel(const float* __restrict__ x,
               const long long* __restrict__ src,
               const long long* __restrict__ dst,
               const float* __restrict__ w,
               float* __restrict__ agg, int nE)
{
    int gid  = blockIdx.x * blockDim.x + threadIdx.x;
    int e    = gid >> 5;
    int lane = gid & 31;
    if (e >= nE) return;

    long long s = src[e];
    long long d = dst[e];
    float    ww = w[e];

    const float4 xs = *(const float4*)(x + (size_t)s * C_DIM + lane * 4);
    const float4 xd = *(const float4*)(x + (size_t)d * C_DIM + lane * 4);

    float4 m;
    m.x = ww * (xs.x - xd.x);
    m.y = ww * (xs.y - xd.y);
    m.z = ww * (xs.z - xd.z);
    m.w = ww * (xs.w - xd.w);

    float* ap = agg + (size_t)d * C_DIM + lane * 4;
    atomicAdd(ap + 0, m.x);
    atomicAdd(ap + 1, m.y);
    atomicAdd(ap + 2, m.z);
    atomicAdd(ap + 3, m.w);
}

// ---------------------------------------------------------------------
// Kernel 4: out = agg @ W + b  (+ optional fused relu + residual)
// Full-precision f32 WMMA: V_WMMA_F32_16X16X4_F32, D(16x16)=A(16x4)B(4x16)+C.
// One wave per 16-row M-tile; it owns all 8 N-tiles (8 independent acc
// chains -> hides WMMA->WMMA RAW latency; A fragment reused 8x per k-step).
//
// Fragment layouts per CDNA5 ISA 7.12.2 (wave32):
//   A 16x4 f32 : lanes 0-15 M=lane, v0=K0 v1=K1 ; lanes 16-31 v0=K2 v1=K3
//   B 4x16 f32 : lanes 0-15 N=lane, v0=B[K0][n] v1=B[K1][n]; lanes16-31 K2/K3
//   C/D 16x16  : lanes 0-15 N=lane, vgpr r = M=r ; lanes 16-31 M=8+r
// ---------------------------------------------------------------------
__global__ void __launch_bounds__(256)
gemm_kernel(const float* __restrict__ agg,
            const float* __restrict__ W,      // [128,128] row-major (k, j)
            const float* __restrict__ bias,   // [128]
            const float* __restrict__ resid,  // [N,128] (layer input)
            float* __restrict__ out,          // [N,128]
            int fuse_relu_res)
{
    const int lane  = threadIdx.x;                 // 0..31
    const int mtile = blockIdx.x * 8 + threadIdx.y;
    if (mtile >= M_TILES) return;                  // whole wave exits together

    const int mb   = mtile * 16;
    const int half = lane >> 4;                    // 0: K0/K1, 1: K2/K3
    const int r    = lane & 15;

    v8f acc[8] = {};                               // 8 N-tiles, 64 VGPRs

    const float* arow = agg + (size_t)(mb + r) * C_DIM;

    #pragma unroll 2
    for (int k = 0; k < C_DIM; k += 4) {
        // A fragment: two contiguous f32 per lane -> single b64 load
        v2f a = *(const v2f*)(arow + k + 2 * half);
        #pragma unroll
        for (int n = 0; n < 8; ++n) {
            const float* bp = W + (size_t)(k + 2 * half) * C_DIM + n * 16 + r;
            v2f b;
            b.x = bp[0];
            b.y = bp[C_DIM];
            acc[n] = __builtin_amdgcn_wmma_f32_16x16x4_f32(
                /*neg_a=*/false, a, /*neg_b=*/false, b,
                /*c_mod=*/(short)0, acc[n],
                /*reuse_a=*/false, /*reuse_b=*/false);
        }
    }

    // Epilogue: bias (+ relu + residual), write out
    #pragma unroll
    for (int n = 0; n < 8; ++n) {
        const int col = n * 16 + r;
        const float bb = bias[col];
        #pragma unroll
        for (int v = 0; v < 8; ++v) {
            const int row = mb + v + 8 * half;
            float val = acc[n][v] + bb;
            if (fuse_relu_res)
                val = fmaxf(val, 0.0f) + resid[(size_t)row * C_DIM + col];
            out[(size_t)row * C_DIM + col] = val;
        }
    }
}

// ---------------------------------------------------------------------
// Host-side orchestration
// ---------------------------------------------------------------------
extern "C" void kernel_launch(void* const* d_in, const int* in_sizes, int n_in,
                              void* d_out, int out_size, void* d_ws, size_t ws_size,
                              hipStream_t stream)
{
    (void)in_sizes; (void)n_in; (void)out_size; (void)ws_size;

    const float*     x      = (const float*)d_in[0];
    const long long* eidx   = (const long long*)d_in[1];   // int64 in reference
    const float*     eattr  = (const float*)d_in[2];
    const float*     lin_w  = (const float*)d_in[3];       // [3,128,128]
    const float*     lin_b  = (const float*)d_in[4];       // [3,128]
    const float*     emlp_w = (const float*)d_in[5];       // [3,8]
    const float*     emlp_b = (const float*)d_in[6];       // [3]

    const long long* src = eidx;            // edge_index[0]
    const long long* dst = eidx + N_EDGES;  // edge_index[1]

    // workspace layout (bytes): agg | w | xA | xB   (total 80,000,000 B)
    char*  ws   = (char*)d_ws;
    float* agg  = (float*)(ws + 0);
    float* wbuf = (float*)(ws + 25600000);   // 50000*128*4
    float* xA   = (float*)(ws + 28800000);   // + 800000*4
    float* xB   = (float*)(ws + 54400000);   // + 50000*128*4

    const int n4   = N_NODES * C_DIM / 4;    // 1,600,000 float4s in agg
    const int ewBlocks   = (N_EDGES + 255) / 256;
    const int zBlocks    = (n4 + 255) / 256;
    const int scBlocks   = (N_EDGES * 32 + 255) / 256;   // wave per edge
    const int gemmBlocks = (M_TILES + 7) / 8;            // 8 waves/block

    for (int layer = 0; layer < N_LAYERS; ++layer) {
        const float* xin  = (layer == 0) ? x  : ((layer == 1) ? xA : xB);
        float*       xout = (layer == 0) ? xA : ((layer == 1) ? xB : (float*)d_out);

        edge_weight_kernel<<<ewBlocks, 256, 0, stream>>>(
            eattr, emlp_w + layer * EDGE_DIM, emlp_b + layer, wbuf, N_EDGES);

        zero_kernel<<<zBlocks, 256, 0, stream>>>((float4*)agg, n4);

        scatter_kernel<<<scBlocks, 256, 0, stream>>>(
            xin, src, dst, wbuf, agg, N_EDGES);

        gemm_kernel<<<gemmBlocks, dim3(32, 8), 0, stream>>>(
            agg, lin_w + (size_t)layer * C_DIM * C_DIM, lin_b + layer * C_DIM,
            xin, xout, (layer < N_LAYERS - 1) ? 1 : 0);
    }
}